// QuLinear_60129542144050
// MI455X (gfx1250) — compile-verified
//
#include <hip/hip_runtime.h>
#include <hip/hip_bf16.h>

// ---------------------------------------------------------------------------
// Strategy (MI455X / gfx1250, wave32):
//   circuit(x) = U @ psi where U is a fixed 1024x1024 complex unitary.
//   psi is real (amplitude encoding of real x) -> Psi = X @ Ure^T and X @ Uim^T:
//   two FP32 GEMMs done with v_wmma_f32_16x16x4_f32.
//   Pauli expectations are fused in the GEMM epilogue from an LDS-resident
//   Psi tile; normalization folded in as a quadratic output scale.
// ---------------------------------------------------------------------------

#define NQ   10
#define DIM  1024          // 2^NQ
#define NLAY 5
#define BATCH 16384
#define XSTRIDE 1028       // padded LDS stride (bank spread: m*1028 % 64 = 4m)

typedef __attribute__((ext_vector_type(2))) float v2f;
typedef __attribute__((ext_vector_type(8))) float v8f;

// ---------------------------------------------------------------------------
// Kernel 1: build W[c][i] = U[i][c]  (row c = circuit applied to basis state c)
// One block per basis state; state lives in 8KB of LDS.
// Qubit q acts on bit (9-q) of the flat index (C-order reshape in reference).
// ---------------------------------------------------------------------------
__global__ __launch_bounds__(256) void build_u_kernel(
    const float* __restrict__ weight,   // [6][30]
    float* __restrict__ wre,            // [DIM][DIM]
    float* __restrict__ wim)            // [DIM][DIM]
{
    __shared__ float sre[DIM];
    __shared__ float sim[DIM];
    const int tid = threadIdx.x;
    const int col = blockIdx.x;

    for (int i = tid; i < DIM; i += 256) {
        sre[i] = (i == col) ? 1.0f : 0.0f;
        sim[i] = 0.0f;
    }
    __syncthreads();

    for (int layer = 0; layer <= NLAY; ++layer) {
        for (int q = 0; q < NQ; ++q) {
            const int mask = 1 << (9 - q);
            const float hx = weight[layer * 30 + 3 * q + 0] * 0.5f;
            const float hy = weight[layer * 30 + 3 * q + 1] * 0.5f;
            const float hz = weight[layer * 30 + 3 * q + 2] * 0.5f;

            // RX: [[c, -i s],[-i s, c]]
            {
                const float c = cosf(hx), s = sinf(hx);
                for (int i = tid; i < DIM; i += 256) {
                    if (!(i & mask)) {
                        const int j = i | mask;
                        const float a0r = sre[i], a0i = sim[i];
                        const float a1r = sre[j], a1i = sim[j];
                        sre[i] = c * a0r + s * a1i;
                        sim[i] = c * a0i - s * a1r;
                        sre[j] = c * a1r + s * a0i;
                        sim[j] = c * a1i - s * a0r;
                    }
                }
                __syncthreads();
            }
            // RY: [[c, -s],[s, c]] (real rotation on both re/im planes)
            {
                const float c = cosf(hy), s = sinf(hy);
                for (int i = tid; i < DIM; i += 256) {
                    if (!(i & mask)) {
                        const int j = i | mask;
                        const float a0r = sre[i], a0i = sim[i];
                        const float a1r = sre[j], a1i = sim[j];
                        sre[i] = c * a0r - s * a1r;
                        sim[i] = c * a0i - s * a1i;
                        sre[j] = s * a0r + c * a1r;
                        sim[j] = s * a0i + c * a1i;
                    }
                }
                __syncthreads();
            }
            // RZ: diag(e^{-ih}, e^{+ih})
            {
                const float c = cosf(hz), s = sinf(hz);
                for (int i = tid; i < DIM; i += 256) {
                    if (!(i & mask)) {
                        const int j = i | mask;
                        const float a0r = sre[i], a0i = sim[i];
                        const float a1r = sre[j], a1i = sim[j];
                        sre[i] = c * a0r + s * a0i;
                        sim[i] = c * a0i - s * a0r;
                        sre[j] = c * a1r - s * a1i;
                        sim[j] = c * a1i + s * a1r;
                    }
                }
                __syncthreads();
            }
        }
        if (layer < NLAY) {
            for (int q = 0; q < NQ; ++q) {
                const int cm = 1 << (9 - q);
                const int tm = 1 << (9 - ((q + 1) % NQ));
                for (int i = tid; i < DIM; i += 256) {
                    if ((i & cm) && !(i & tm)) {
                        const int j = i | tm;
                        const float tr = sre[i], ti = sim[i];
                        sre[i] = sre[j]; sim[i] = sim[j];
                        sre[j] = tr;     sim[j] = ti;
                    }
                }
                __syncthreads();
            }
        }
    }

    // W row c = U column c (so GEMM is X[b, :] @ W -> Psi[b, :])
    for (int i = tid; i < DIM; i += 256) {
        wre[(size_t)col * DIM + i] = sre[i];
        wim[(size_t)col * DIM + i] = sim[i];
    }
}

// ---------------------------------------------------------------------------
// Kernel 2: per-row quadratic normalization scale = 1 / max(||x||, 1e-12)^2
// ---------------------------------------------------------------------------
__global__ __launch_bounds__(256) void rownorm_kernel(
    const float* __restrict__ x, float* __restrict__ scale)
{
    const int row  = blockIdx.x * 8 + (threadIdx.x >> 5);
    const int lane = threadIdx.x & 31;
    const float* xr = x + (size_t)row * DIM;
    float s = 0.0f;
    for (int i = lane; i < DIM; i += 32) {
        const float v = xr[i];
        s += v * v;
    }
    #pragma unroll
    for (int off = 16; off; off >>= 1) s += __shfl_xor(s, off, 32);
    if (lane == 0) scale[row] = 1.0f / fmaxf(s, 1e-24f);
}

// ---------------------------------------------------------------------------
// Kernel 3: fused FP32 WMMA GEMM (Psi = X @ W, re & im) + Pauli expectations.
// Block = 16 batch rows x 1024 columns; 8 waves, each wave owns 128 columns.
// ---------------------------------------------------------------------------
__global__ __launch_bounds__(256) void gemm_expect_kernel(
    const float* __restrict__ x,       // [BATCH][DIM]
    const float* __restrict__ wre,     // [DIM][DIM]
    const float* __restrict__ wim,     // [DIM][DIM]
    const float* __restrict__ scale,   // [BATCH]
    float* __restrict__ out)           // [BATCH][30]
{
    extern __shared__ float smem[];    // 32768 floats = 128KB (X tile, then Psi)
    const int tid  = threadIdx.x;
    const int lane = tid & 31;
    const int wv   = tid >> 5;
    const int r0   = blockIdx.x * 16;

    // ---- stage 16 contiguous rows of X into LDS (padded stride) ----
    {
        const float4* src = (const float4*)(x + (size_t)r0 * DIM);
        for (int idx = tid; idx < (16 * DIM) / 4; idx += 256) {
            const float4 v = src[idx];
            const int flat = idx * 4;
            const int m  = flat >> 10;
            const int kk = flat & (DIM - 1);
            float* dst = &smem[m * XSTRIDE + kk];
            dst[0] = v.x; dst[1] = v.y; dst[2] = v.z; dst[3] = v.w;
        }
    }
    __syncthreads();

    const int m    = lane & 15;        // A-frag row (both half-waves)
    const int half = lane >> 4;        // 0: K=k,k+1  1: K=k+2,k+3
    const int nbase = wv * 128 + (lane & 15);

    const v8f vzero = {0.f, 0.f, 0.f, 0.f, 0.f, 0.f, 0.f, 0.f};
    v8f accR[8], accI[8];
    #pragma unroll
    for (int t = 0; t < 8; ++t) { accR[t] = vzero; accI[t] = vzero; }

    // ---- K loop: 256 steps of 16x16x4 FP32 WMMA over 8 N-tiles (re+im) ----
    for (int k = 0; k < DIM; k += 4) {
        const int ka = k + 2 * half;
        v2f a;
        a.x = smem[m * XSTRIDE + ka];
        a.y = smem[m * XSTRIDE + ka + 1];
        const float* wr0 = wre + (size_t)ka * DIM;
        const float* wi0 = wim + (size_t)ka * DIM;
        #pragma unroll
        for (int t = 0; t < 8; ++t) {
            const int col = nbase + t * 16;
            v2f br; br.x = wr0[col]; br.y = wr0[DIM + col];
            accR[t] = __builtin_amdgcn_wmma_f32_16x16x4_f32(
                false, a, false, br, (short)0, accR[t], false, false);
            v2f bi; bi.x = wi0[col]; bi.y = wi0[DIM + col];
            accI[t] = __builtin_amdgcn_wmma_f32_16x16x4_f32(
                false, a, false, bi, (short)0, accI[t], false, false);
        }
    }
    __syncthreads();   // X tile dead; reuse LDS for the Psi tile

    float* yre = smem;               // [16][DIM]
    float* yim = smem + 16 * DIM;    // [16][DIM]
    #pragma unroll
    for (int t = 0; t < 8; ++t) {
        const int col = nbase + t * 16;
        #pragma unroll
        for (int v = 0; v < 8; ++v) {
            const int r = v + 8 * half;   // ISA C/D layout: row = vgpr + 8*(lane>=16)
            yre[r * DIM + col] = accR[t][v];
            yim[r * DIM + col] = accI[t][v];
        }
    }
    __syncthreads();

    // ---- Pauli expectations: wave handles 2 rows; outputs [X0..9,Y0..9,Z0..9]
    for (int rr = 0; rr < 2; ++rr) {
        const int r = wv * 2 + rr;
        const int b = r0 + r;
        float acc[30];
        #pragma unroll
        for (int o = 0; o < 30; ++o) acc[o] = 0.0f;

        const float* pr = yre + r * DIM;
        const float* pi = yim + r * DIM;
        for (int i = lane; i < DIM; i += 32) {
            const float ar = pr[i], ai = pi[i];
            const float p = ar * ar + ai * ai;
            #pragma unroll
            for (int q = 0; q < NQ; ++q) {
                const int mask = 1 << (9 - q);
                if (i & mask) {
                    acc[20 + q] -= p;                    // Z: -|psi_i|^2
                } else {
                    acc[20 + q] += p;                    // Z: +|psi_i|^2
                    const int j = i | mask;
                    const float br = pr[j], bi = pi[j];
                    acc[q]      += 2.0f * (ar * br + ai * bi);  // X
                    acc[10 + q] += 2.0f * (ar * bi - ai * br);  // Y
                }
            }
        }

        const float sc = scale[b];
        #pragma unroll
        for (int o = 0; o < 30; ++o) {
            float v = acc[o];
            #pragma unroll
            for (int off = 16; off; off >>= 1) v += __shfl_xor(v, off, 32);
            if (lane == 0) out[(size_t)b * 30 + o] = v * sc;
        }
    }
}

// ---------------------------------------------------------------------------
extern "C" void kernel_launch(void* const* d_in, const int* in_sizes, int n_in,
                              void* d_out, int out_size, void* d_ws, size_t ws_size,
                              hipStream_t stream) {
    const float* x      = (const float*)d_in[0];   // [16384][1024] f32
    const float* weight = (const float*)d_in[1];   // [6][30] f32
    float* out = (float*)d_out;                    // [16384][30] f32

    float* wre   = (float*)d_ws;                   // 4 MB
    float* wim   = wre + (size_t)DIM * DIM;        // 4 MB
    float* scale = wim + (size_t)DIM * DIM;        // 64 KB

    build_u_kernel<<<DIM, 256, 0, stream>>>(weight, wre, wim);
    rownorm_kernel<<<BATCH / 8, 256, 0, stream>>>(x, scale);
    gemm_expect_kernel<<<BATCH / 16, 256, 32768 * sizeof(float), stream>>>(
        x, wre, wim, scale, out);
}